// GCNLayer_27986006901492
// MI455X (gfx1250) — compile-verified
//
#include <hip/hip_runtime.h>

// ---------------------------------------------------------------------------
// GCN layer, fused:  out[e] = 0.5*(h2W[src[e]] + h2W[dst[e]]) + b
// where h2W = (segment_sum(segment_mean(edge_feats,dst)[src], dst)) @ W.T
//
// MI455X reasoning:
//  - Streaming traffic (edge_feats read-once 307MB, out write-once 307MB) uses
//    non-temporal hints so the L2 (192MB) keeps the node tables (~19MB) that
//    the scatter/gather kernels reuse heavily.
//  - Node-level GEMM (50000x48 @ 48x48) runs on V_WMMA_F32_16X16X4_F32;
//    50000 = 3125*16 and 48 = 3*16, so tiling is exact and EXEC is all-ones
//    for every WMMA wave (ISA 7.12 requirement).
//  - Hoisting the 48x48 linear from per-edge to per-node cuts GEMM FLOPs 32x.
// ---------------------------------------------------------------------------

#define NNODES 50000
#define FDIM   48

typedef __attribute__((ext_vector_type(2))) float v2f;
typedef __attribute__((ext_vector_type(4))) float v4f;
typedef __attribute__((ext_vector_type(8))) float v8f;

// ---- kernel 0: zero workspace region (deg | nodesum | h2), b128 stores -----
__global__ void k_zero(v4f* __restrict__ p, int n4) {
    int i = blockIdx.x * blockDim.x + threadIdx.x;
    if (i < n4) p[i] = (v4f){0.f, 0.f, 0.f, 0.f};
}

// ---- kernel 1: scatter-add edge feats into nodesum, count degree -----------
// edge_feats is read exactly once -> non-temporal load (TH=NT).
__global__ void k_scatter_mean(const float* __restrict__ ef,
                               const int* __restrict__ dst,
                               float* __restrict__ deg,
                               float* __restrict__ nodesum,
                               int total) {
    int tid = blockIdx.x * blockDim.x + threadIdx.x;
    if (tid >= total) return;
    int e = tid / FDIM;
    int f = tid - e * FDIM;
    int d = dst[e];
    float v = __builtin_nontemporal_load(ef + tid);
    atomicAdd(&nodesum[d * FDIM + f], v);
    if (f == 0) atomicAdd(&deg[d], 1.0f);
}

// ---- kernel 2: node_h = nodesum / max(deg,1)   (in place) ------------------
__global__ void k_normalize(float* __restrict__ nodeh,
                            const float* __restrict__ deg,
                            int total) {
    int tid = blockIdx.x * blockDim.x + threadIdx.x;
    if (tid >= total) return;
    int n = tid / FDIM;
    nodeh[tid] = nodeh[tid] / fmaxf(deg[n], 1.0f);
}

// ---- kernel 3: h2[dst] += node_h[src]  (all L2-resident) -------------------
__global__ void k_scatter_sum(const int* __restrict__ src,
                              const int* __restrict__ dst,
                              const float* __restrict__ nodeh,
                              float* __restrict__ h2,
                              int total) {
    int tid = blockIdx.x * blockDim.x + threadIdx.x;
    if (tid >= total) return;
    int e = tid / FDIM;
    int f = tid - e * FDIM;
    atomicAdd(&h2[dst[e] * FDIM + f], nodeh[src[e] * FDIM + f]);
}

// ---- kernel 4: h2W = h2 @ W.T  via V_WMMA_F32_16X16X4_F32 ------------------
// One wave computes a 16(node) x 16(outfeat) tile, K=48 in 12 steps of 4.
// A (16x4 f32): lanes 0-15 -> M=lane, {K=k0,k0+1}; lanes 16-31 -> M=lane-16,
//               {K=k0+2,k0+3}   (ISA 7.12.2 "32-bit A-Matrix 16x4")
// B (4x16 f32): mirrored striping: v0 = B[k0+koff, n], v1 = B[k0+koff+1, n]
//               with B[k,n] = W[n0+n, k]  (W is [out,in], row-major)
// C/D (16x16 f32, 8 VGPRs): c[i] = D[M = i + 8*(lane>=16), N = lane&15]
__global__ void k_node_gemm(const float* __restrict__ h2,
                            const float* __restrict__ W,
                            float* __restrict__ h2W,
                            int nTiles /* = (NNODES/16)*3 */) {
    int lane = threadIdx.x & 31;
    int wave = (blockIdx.x * blockDim.x + threadIdx.x) >> 5;
    if (wave >= nTiles) return;          // wave-uniform: EXEC stays all-ones

    int mt = wave / 3;                   // node tile
    int nt = wave - mt * 3;              // out-feature tile
    int m0 = mt << 4;
    int n0 = nt << 4;

    int half = lane >> 4;                // 0: lanes 0-15, 1: lanes 16-31
    int mrow = lane & 15;
    int ncol = lane & 15;
    int koff = half << 1;                // 0 or 2

    const float* arow = h2 + (m0 + mrow) * FDIM + koff;
    const float* brow = W + (n0 + ncol) * FDIM + koff;

    v8f c = {};
    #pragma unroll
    for (int k0 = 0; k0 < FDIM; k0 += 4) {
        v2f a; a.x = arow[k0]; a.y = arow[k0 + 1];
        v2f b; b.x = brow[k0]; b.y = brow[k0 + 1];
        c = __builtin_amdgcn_wmma_f32_16x16x4_f32(
                /*neg_a=*/false, a, /*neg_b=*/false, b,
                /*c_mod=*/(short)0, c, /*reuse_a=*/false, /*reuse_b=*/false);
    }

    int mbase = m0 + (half << 3);
    float* outp = h2W + mbase * FDIM + n0 + ncol;
    #pragma unroll
    for (int i = 0; i < 8; ++i)
        outp[i * FDIM] = c[i];
}

// ---- kernel 5: out[e,4f..4f+3] = 0.5*(h2W[src]+h2W[dst]) + b ---------------
// One thread per 16B quarter-row: b128 gathers from L2-resident h2W,
// b128 non-temporal store of the 307MB output (write-once, never re-read).
__global__ void k_edge_out(const int* __restrict__ src,
                           const int* __restrict__ dst,
                           const float* __restrict__ h2W,
                           const float* __restrict__ bias,
                           float* __restrict__ out,
                           int total4 /* = E * 12 */) {
    int tid = blockIdx.x * blockDim.x + threadIdx.x;
    if (tid >= total4) return;
    int e = tid / 12;
    int q = tid - e * 12;                 // quarter-row index, 4 floats each
    int fo = q << 2;
    const v4f* ps = (const v4f*)(h2W + src[e] * FDIM + fo);
    const v4f* pd = (const v4f*)(h2W + dst[e] * FDIM + fo);
    const v4f* pb = (const v4f*)(bias + fo);
    v4f r = 0.5f * (*ps + *pd) + *pb;
    __builtin_nontemporal_store(r, (v4f*)(out + e * FDIM + fo));
}

// ---------------------------------------------------------------------------
extern "C" void kernel_launch(void* const* d_in, const int* in_sizes, int n_in,
                              void* d_out, int out_size, void* d_ws, size_t ws_size,
                              hipStream_t stream) {
    const float* edge_feats = (const float*)d_in[0];   // [E,48]
    const int*   src        = (const int*)d_in[1];     // [E]
    const int*   dst        = (const int*)d_in[2];     // [E]
    const float* W          = (const float*)d_in[3];   // [48,48] row-major [out,in]
    const float* b          = (const float*)d_in[4];   // [48]
    float*       out        = (float*)d_out;           // [E,48]

    const int E = in_sizes[1];

    // workspace layout (floats): deg[N] | buf1[N*48] | buf2[N*48]
    // N*1 = 50000 floats = 200000 B (16B-aligned), rows are 192 B.
    float* deg  = (float*)d_ws;
    float* buf1 = deg + NNODES;            // nodesum -> node_h -> h2W
    float* buf2 = buf1 + NNODES * FDIM;    // h2
    const int wsFloats = NNODES * (1 + 2 * FDIM);   // divisible by 4

    const int BT = 256;
    const int edgeTotal = E * FDIM;
    const int nodeTotal = NNODES * FDIM;

    // 0) zero accumulators (must happen every call: graph replay)
    const int n4 = wsFloats / 4;
    k_zero<<<(n4 + BT - 1) / BT, BT, 0, stream>>>((v4f*)d_ws, n4);

    // 1) nodesum[dst] += edge_feats ; deg[dst] += 1
    k_scatter_mean<<<(edgeTotal + BT - 1) / BT, BT, 0, stream>>>(
        edge_feats, dst, deg, buf1, edgeTotal);

    // 2) node_h = nodesum / max(deg,1)
    k_normalize<<<(nodeTotal + BT - 1) / BT, BT, 0, stream>>>(buf1, deg, nodeTotal);

    // 3) h2[dst] += node_h[src]
    k_scatter_sum<<<(edgeTotal + BT - 1) / BT, BT, 0, stream>>>(
        src, dst, buf1, buf2, edgeTotal);

    // 4) h2W = h2 @ W.T   (WMMA; overwrites buf1)
    const int nTiles = (NNODES / 16) * 3;              // 9375 waves
    const int wavesPerBlock = BT / 32;                 // 8
    k_node_gemm<<<(nTiles + wavesPerBlock - 1) / wavesPerBlock, BT, 0, stream>>>(
        buf2, W, buf1, nTiles);

    // 5) out = 0.5*(h2W[src]+h2W[dst]) + b   (b128, NT stores)
    const int total4 = E * 12;
    k_edge_out<<<(total4 + BT - 1) / BT, BT, 0, stream>>>(
        src, dst, buf1, b, out, total4);
}